// HydraBackbonePlus_22505628631802
// MI455X (gfx1250) — compile-verified
//
#include <hip/hip_runtime.h>

typedef float v2f __attribute__((ext_vector_type(2)));
typedef float v8f __attribute__((ext_vector_type(8)));

#define C_IN_    12
#define SEQ_     8192
#define K_       8
#define G_       32      // _G = G/DIVISOR
#define CPER_    6
#define NBR_     20      // 10 dilations * 2 (x / diff_x)
#define OUTCH_   1280    // NBR_ * 2 * G_
#define WAVES_   4
#define BLOCK_   (WAVES_ * 32)
#define HALO_    2048    // 4 * d_max (d_max = 512)
#define TOTAL_   (HALO_ + SEQ_ + HALO_ + 16)   // +16 slack for last partial tile

__global__ __launch_bounds__(BLOCK_) void hydra_backbone_kernel(
    const float* __restrict__ X,   // (32, 12, 8192)
    const float* __restrict__ W,   // (10, 2, 256, 1, 9)
    const int*   __restrict__ I,   // (10, 2, 32, 6)
    float* __restrict__ out)       // (32, 1280, 8)
{
  __shared__ float s_halo[TOTAL_];  // zero halo + staged gathered signal (~48 KB)
  __shared__ float s_cmax[K_];
  __shared__ float s_cmin[K_];

  const int tid  = threadIdx.x;
  const int lane = tid & 31;
  const int wave = tid >> 5;

  const int g  = blockIdx.x & (G_ - 1);
  const int b  = (blockIdx.x >> 5) & 31;
  const int br = blockIdx.x >> 10;         // 0..19 : br = di*2 + j
  const int di = br >> 1;
  const int j  = br & 1;
  const int d  = 1 << di;
  const int L  = (j == 0) ? SEQ_ : (SEQ_ - 1);

  if (tid < K_) { s_cmax[tid] = 0.f; s_cmin[tid] = 0.f; }

  // zero both halos (conv zero-padding lives here; no bounds checks later)
  for (int t = tid; t < HALO_; t += BLOCK_) s_halo[t] = 0.f;
  for (int t = tid; t < TOTAL_ - (HALO_ + L); t += BLOCK_) s_halo[HALO_ + L + t] = 0.f;

  // channel-gather indices for this (branch, group)
  int cidx[CPER_];
  {
    const int* Ip = I + ((size_t)br * G_ + g) * CPER_;
    #pragma unroll
    for (int m = 0; m < CPER_; ++m) cidx[m] = Ip[m];
  }

  // Stage inp[t] = sum_m src[cidx[m]][t] into LDS (src = X or diff(X))
  const float* Xb = X + (size_t)b * (C_IN_ * SEQ_);
  for (int t = tid; t < L; t += BLOCK_) {
    float s = 0.f;
    #pragma unroll
    for (int m = 0; m < CPER_; ++m) {
      const float* row = Xb + cidx[m] * SEQ_;
      __builtin_prefetch(row + t + BLOCK_, 0, 0);
      const float v0 = row[t];
      s += (j == 0) ? v0 : (row[t + 1] - v0);
    }
    s_halo[HALO_ + t] = s;
  }
  __syncthreads();

  // Per-lane A fragments (weights), 3 tap-chunks of the 16x4 f32 A layout:
  // lane l: M = l&15 (rows >=8 zero); VGPR0 = tap r0, VGPR1 = tap r0+1,
  // r0 = 4*chunk + 2*laneHalf. Taps > 8 get zero weights.
  const float* Wd = W + ((size_t)br * (K_ * G_) + (size_t)g * K_) * 9;
  const int m  = lane & 15;
  const int hi = (lane >> 4) & 1;
  v2f a[3];
  #pragma unroll
  for (int cc = 0; cc < 3; ++cc) {
    const int r0 = 4 * cc + 2 * hi;
    a[cc].x = (m < K_ && r0 < 9)       ? Wd[m * 9 + r0]     : 0.f;
    a[cc].y = (m < K_ && (r0 + 1) < 9) ? Wd[m * 9 + r0 + 1] : 0.f;
  }

  // Per-lane constant LDS pointers for the 6 B elements. For taps > 8 the
  // A-weight is zero, so clamp the tap to 8: any valid value works.
  const int n = lane & 15;                 // N (position within 16-wide tile)
  const float* bp[6];
  #pragma unroll
  for (int cc = 0; cc < 3; ++cc) {
    int r0 = 4 * cc + 2 * hi;
    int r1 = r0 + 1;
    if (r0 > 8) r0 = 8;
    if (r1 > 8) r1 = 8;
    bp[2 * cc]     = s_halo + HALO_ + n + (r0 - 4) * d;
    bp[2 * cc + 1] = s_halo + HALO_ + n + (r1 - 4) * d;
  }

  float accMax[K_], accMin[K_];
  #pragma unroll
  for (int k = 0; k < K_; ++k) { accMax[k] = 0.f; accMin[k] = 0.f; }

  const int nChunks = (L + 15) >> 4;
  for (int ci = wave; ci < nChunks; ci += WAVES_) {
    const int t0 = ci << 4;

    // 6 straight-line LDS loads (no bounds checks), then 3 chained WMMAs
    v2f b0, b1, b2;
    b0.x = bp[0][t0];  b0.y = bp[1][t0];
    b1.x = bp[2][t0];  b1.y = bp[3][t0];
    b2.x = bp[4][t0];  b2.y = bp[5][t0];

    v8f c = {};
    c = __builtin_amdgcn_wmma_f32_16x16x4_f32(false, a[0], false, b0, (short)0, c, false, false);
    c = __builtin_amdgcn_wmma_f32_16x16x4_f32(false, a[1], false, b1, (short)0, c, false, false);
    c = __builtin_amdgcn_wmma_f32_16x16x4_f32(false, a[2], false, b2, (short)0, c, false, false);

    const int t = t0 + n;
    if (lane < 16 && t < L) {
      // first-occurrence argmax/argmin over the 8 conv channels (matches jnp)
      float mx = c[0], mn = c[0];
      int amax = 0, amin = 0;
      #pragma unroll
      for (int k = 1; k < K_; ++k) {
        const float v = c[k];
        if (v > mx) { mx = v; amax = k; }
        if (v < mn) { mn = v; amin = k; }
      }
      #pragma unroll
      for (int k = 0; k < K_; ++k) {
        accMax[k] += (amax == k) ? mx  : 0.f;
        accMin[k] += (amin == k) ? 1.f : 0.f;
      }
    }
  }

  // wave32 butterfly reduce, then cross-wave combine via LDS float atomics
  #pragma unroll
  for (int k = 0; k < K_; ++k) {
    float vM = accMax[k], vN = accMin[k];
    #pragma unroll
    for (int off = 16; off >= 1; off >>= 1) {
      vM += __shfl_xor(vM, off, 32);
      vN += __shfl_xor(vN, off, 32);
    }
    if (lane == 0) {
      atomicAdd(&s_cmax[k], vM);
      atomicAdd(&s_cmin[k], vN);
    }
  }
  __syncthreads();

  if (tid < K_) {
    const size_t base = (size_t)b * ((size_t)OUTCH_ * K_);
    const float vM = s_cmax[tid];
    const float vN = s_cmin[tid];
    out[base + (size_t)(br * 64 + g) * K_ + tid]      = vM > 0.f ? vM : 0.f;  // cmax block
    out[base + (size_t)(br * 64 + 32 + g) * K_ + tid] = vN > 0.f ? vN : 0.f;  // cmin block
  }
}

extern "C" void kernel_launch(void* const* d_in, const int* in_sizes, int n_in,
                              void* d_out, int out_size, void* d_ws, size_t ws_size,
                              hipStream_t stream) {
  const float* X = (const float*)d_in[0];
  const float* W = (const float*)d_in[1];
  const int*   I = (const int*)d_in[2];
  float* out = (float*)d_out;

  dim3 grid(NBR_ * 32 * G_);   // 20 branches * 32 batch * 32 groups = 20480
  hydra_backbone_kernel<<<grid, BLOCK_, 0, stream>>>(X, W, I, out);
}